// SMorphNetDoubleClassif_90864328114814
// MI455X (gfx1250) — compile-verified
//
#include <hip/hip_runtime.h>
#include <hip/hip_bf16.h>

#define HW   28
#define KF   7
#define PAD  3
#define TILE 34   // 28 + 2*3 halo

typedef __attribute__((ext_vector_type(2))) float v2f;
typedef __attribute__((ext_vector_type(8))) float v8f;

// ---------------------------------------------------------------------------
// SMorph: out(p) = sum_s z_s * softmax_s(alpha*z)  with z_s = x(p+s) + w(s)
// One block per (batch, branch) image. LDS-tiled with zero halo (== SAME pad:
// padded taps contribute z = 0 + w(s), exactly as the reference softmax does).
// ---------------------------------------------------------------------------
__global__ void smorph_kernel(const float* __restrict__ in, long in_branch_stride,
                              const float* __restrict__ sm_w,
                              const float* __restrict__ sm_alpha,
                              int pass, float* __restrict__ out)
{
    const int b  = blockIdx.x;   // 0..511
    const int br = blockIdx.y;   // 0..7
    const int tid = threadIdx.x;

    __shared__ float tile[TILE * TILE];
    __shared__ float wsh[KF * KF];

    const float* img = in + (long)br * in_branch_stride + (long)b * (HW * HW);

    for (int i = tid; i < TILE * TILE; i += blockDim.x) {
        int r = i / TILE - PAD;
        int c = i % TILE - PAD;
        float v = 0.0f;
        if ((unsigned)r < (unsigned)HW && (unsigned)c < (unsigned)HW)
            v = img[r * HW + c];
        tile[i] = v;
    }
    if (tid < KF * KF)
        wsh[tid] = sm_w[(br * 2 + pass) * (KF * KF) + tid];
    __syncthreads();

    const float alpha = sm_alpha[br * 2 + pass];   // uniform -> scalar load
    float* o = out + ((long)br * 512 + b) * (HW * HW);

    for (int p = tid; p < HW * HW; p += blockDim.x) {
        const int py = p / HW, px = p % HW;
        const float* t0 = &tile[py * TILE + px];

        float z[KF * KF];
        float tmax = -3.402823466e38f;
#pragma unroll
        for (int dy = 0; dy < KF; ++dy) {
#pragma unroll
            for (int dx = 0; dx < KF; ++dx) {
                float zz = t0[dy * TILE + dx] + wsh[dy * KF + dx];
                z[dy * KF + dx] = zz;
                tmax = fmaxf(tmax, alpha * zz);
            }
        }
        float num = 0.0f, den = 0.0f;
#pragma unroll
        for (int s = 0; s < KF * KF; ++s) {
            float e = __expf(fmaf(alpha, z[s], -tmax));
            num = fmaf(z[s], e, num);
            den += e;
        }
        o[p] = num / den;
    }
}

// ---------------------------------------------------------------------------
// Double 2x2 avg-pool == 4x4 mean, fused with branch-major feature repack:
// feat[b, br*49 + r*7 + c] = mean_{4x4}( y2[br][b] )
// ---------------------------------------------------------------------------
__global__ void pool_pack_kernel(const float* __restrict__ y,
                                 float* __restrict__ feat)
{
    int idx = blockIdx.x * blockDim.x + threadIdx.x;
    if (idx >= 512 * 392) return;
    int b  = idx / 392;
    int f  = idx - b * 392;
    int br = f / 49;
    int s  = f - br * 49;
    int r  = s / 7, c = s - r * 7;

    const float* src = y + ((long)br * 512 + b) * (HW * HW) + (r * 4) * HW + (c * 4);
    float sum = 0.0f;
#pragma unroll
    for (int i = 0; i < 4; ++i)
#pragma unroll
        for (int j = 0; j < 4; ++j)
            sum += src[i * HW + j];
    feat[idx] = sum * 0.0625f;
}

// ---------------------------------------------------------------------------
// out[m,n] = sigmoid( sum_k A[m,k]*W[n,k] + bias[n] )  via V_WMMA_F32_16X16X4_F32
// One wave32 per 16x16 output tile; M=512 (multiple of 16). Padded N columns
// are handled BRANCHLESSLY: load from a clamped (always-valid) W row and
// multiply by a 0/1 mask, so the compiler can speculate + merge loads into
// global_load_b64 instead of emitting exec-masked branchy b32 loads.
// All wave exits are wave-uniform -> EXEC all-ones at every WMMA.
// ---------------------------------------------------------------------------
__global__ void gemm_bias_sigmoid(const float* __restrict__ A,
                                  const float* __restrict__ W,
                                  const float* __restrict__ bias,
                                  float* __restrict__ out,
                                  int N, int K, int tilesN, int totalTiles)
{
    const int wave = (blockIdx.x * blockDim.x + threadIdx.x) >> 5; // uniform per wave
    if (wave >= totalTiles) return;

    const int tm   = wave / tilesN;
    const int tn   = wave - tm * tilesN;
    const int lane = threadIdx.x & 31;
    const int half = lane >> 4;       // k-pair select
    const int lm   = lane & 15;       // M (A) / N (B) index

    const int  arow   = tm * 16 + lm;
    const int  bcol   = tn * 16 + lm;
    const bool bvalid = (bcol < N);
    const int  bcolc  = bvalid ? bcol : (N - 1);   // clamped: loads always in-bounds
    const float bmask = bvalid ? 1.0f : 0.0f;      // zero out padded columns

    const float* ap = A + (long)arow  * K + half * 2;
    const float* wp = W + (long)bcolc * K + half * 2;

    v8f acc = {};
    for (int k = 0; k < K; k += 4) {               // K is a multiple of 4 here
        v2f a, bf;
        a.x  = ap[k];
        a.y  = ap[k + 1];
        bf.x = wp[k]     * bmask;                  // branchless padding
        bf.y = wp[k + 1] * bmask;
        acc = __builtin_amdgcn_wmma_f32_16x16x4_f32(
            /*neg_a=*/false, a, /*neg_b=*/false, bf,
            /*c_mod=*/(short)0, acc, /*reuse_a=*/false, /*reuse_b=*/false);
    }

    if (bvalid) {
        const float bv = bias[bcol];
#pragma unroll
        for (int v = 0; v < 8; ++v) {
            int row = tm * 16 + half * 8 + v;      // C/D layout: M = vgpr + 8*(lane>=16)
            float x = acc[v] + bv;
            out[(long)row * N + bcol] = 1.0f / (1.0f + __expf(-x));
        }
    }
}

// ---------------------------------------------------------------------------
extern "C" void kernel_launch(void* const* d_in, const int* in_sizes, int n_in,
                              void* d_out, int out_size, void* d_ws, size_t ws_size,
                              hipStream_t stream)
{
    const float* x        = (const float*)d_in[0];
    const float* sm_w     = (const float*)d_in[1];
    const float* sm_alpha = (const float*)d_in[2];
    const float* W1       = (const float*)d_in[3];
    const float* b1       = (const float*)d_in[4];
    const float* W2       = (const float*)d_in[5];
    const float* b2       = (const float*)d_in[6];
    const float* W3       = (const float*)d_in[7];
    const float* b3       = (const float*)d_in[8];
    float* outp = (float*)d_out;

    const size_t imgElems = (size_t)8 * 512 * HW * HW;   // 3,211,264 floats
    float* bufA = (float*)d_ws;
    float* bufB = bufA + imgElems;
    // bufA is dead after SMorph pass 2 -> reuse it for MLP activations
    float* feat = bufA;                    // [512, 392]
    float* h1   = feat + 512 * 392;        // [512, 120]
    float* h2   = h1   + 512 * 120;        // [512, 84]

    dim3 gridS(512, 8);
    smorph_kernel<<<gridS, 256, 0, stream>>>(x,    (long)0,           sm_w, sm_alpha, 0, bufA);
    smorph_kernel<<<gridS, 256, 0, stream>>>(bufA, (long)(512*HW*HW), sm_w, sm_alpha, 1, bufB);

    pool_pack_kernel<<<(512 * 392 + 255) / 256, 256, 0, stream>>>(bufB, feat);

    // GEMM1: 512x392 -> 120   (32 x 8 tiles = 256 waves)
    gemm_bias_sigmoid<<<256 / 4, 128, 0, stream>>>(feat, W1, b1, h1, 120, 392, 8, 256);
    // GEMM2: 512x120 -> 84    (32 x 6 tiles = 192 waves)
    gemm_bias_sigmoid<<<192 / 4, 128, 0, stream>>>(h1,   W2, b2, h2,  84, 120, 6, 192);
    // GEMM3: 512x84  -> 10    (32 x 1 tiles = 32 waves)
    gemm_bias_sigmoid<<< 32 / 4, 128, 0, stream>>>(h2,   W3, b3, outp, 10,  84, 1,  32);
}